// WeightedTensorProduct_13254269075604
// MI455X (gfx1250) — compile-verified
//
#include <hip/hip_runtime.h>
#include <stdint.h>

// ---------------------------------------------------------------------------
// WeightedTensorProduct (e3nn-style sparse CG contraction) for MI455X/gfx1250.
//
// out[b,mo,c] = sum_k CG[k] * x1[b,M1[k],c] * x2[b,M2[k],c] * w[b,l[k],c]
//   B=4096, NM=25, C=64, NNZ=1000, NW=55, M_out sorted.
//
// Bandwidth-bound: 137 MB HBM floor (~5.9us @ 23.3TB/s); after LDS staging the
// kernel is LDS-operand-read bound (3 gathered reads per (k,c)).  WMMA would
// add 12x dense FLOPs for zero traffic savings, so we instead use the CDNA5
// async global->LDS path (ASYNCcnt), wave-uniform LDS broadcasts for the COO
// metadata, b128 LDS gathers (4 channels/thread), and sorted-segment register
// accumulation flushed with ds_add_f32 atomics.
// ---------------------------------------------------------------------------

typedef float v4f __attribute__((ext_vector_type(4)));

namespace cfg {
constexpr int NM    = 25;
constexpr int C     = 64;
constexpr int NNZ   = 1000;
constexpr int NW    = 55;
constexpr int BLOCK = 256;            // 8 wave32
constexpr int QUADS = 16;             // 16 quads * 4 channels = 64 channels
constexpr int GROUPS = BLOCK / QUADS; // 16 contiguous k-chunks
constexpr int CHUNK = (NNZ + GROUPS - 1) / GROUPS; // 63
}

__device__ __forceinline__ uint32_t lds_off_u32(const void* p) {
  // Low 32 bits of a generic pointer to LDS = workgroup-relative LDS byte
  // offset (flat aperture lives in the high bits) -- exactly what the async
  // instruction's VDST wants.
  return (uint32_t)(uintptr_t)p;
}

// Async DMA: copies 16 bytes per active lane from global base saddr plus offset vaddr to the LDS offset in vdst,
// tracked with ASYNCcnt, no VGPR round-trip.
__device__ __forceinline__ void async_copy_b128(uint32_t lds_byte_off,
                                                unsigned long long gbase,
                                                uint32_t gbyte_off) {
  asm volatile("global_load_async_to_lds_b128 %0, %1, %2"
               :
               : "v"(lds_byte_off), "v"(gbyte_off), "s"(gbase)
               : "memory");
}

__device__ __forceinline__ void wait_asynccnt0() {
#if __has_builtin(__builtin_amdgcn_s_wait_asynccnt)
  __builtin_amdgcn_s_wait_asynccnt(0);
#else
  asm volatile("s_wait_asynccnt 0" ::: "memory");
#endif
}

__global__ __launch_bounds__(cfg::BLOCK)
void wtp_kernel(const float* __restrict__ x1, const float* __restrict__ x2,
                const float* __restrict__ weight, const float* __restrict__ cg,
                const int* __restrict__ m_out, const int* __restrict__ m1,
                const int* __restrict__ m2, const int* __restrict__ l_ind,
                float* __restrict__ out) {
  using namespace cfg;

  __shared__ __align__(16) float s_x1[NM * C];    //  6400 B
  __shared__ __align__(16) float s_x2[NM * C];    //  6400 B
  __shared__ __align__(16) float s_w [NW * C];    // 14080 B
  __shared__ __align__(16) float s_out[NM * C];   //  6400 B
  __shared__ __align__(8)  uint2 s_meta[NNZ];     //  8000 B  {packed idx, cg}

  const int tid = (int)threadIdx.x;
  const int b   = (int)blockIdx.x;

  // ---- Stage tiles for this batch element into LDS via async DMA ----------
  const unsigned long long g1 = (unsigned long long)(uintptr_t)(x1 + (size_t)b * (NM * C));
  const unsigned long long g2 = (unsigned long long)(uintptr_t)(x2 + (size_t)b * (NM * C));
  const unsigned long long gw = (unsigned long long)(uintptr_t)(weight + (size_t)b * (NW * C));
  const uint32_t l1 = lds_off_u32(s_x1);
  const uint32_t l2 = lds_off_u32(s_x2);
  const uint32_t lw = lds_off_u32(s_w);

  for (int i = tid; i < (NM * C) / 4; i += BLOCK) {   // 400 x b128 each
    async_copy_b128(l1 + (uint32_t)i * 16u, g1, (uint32_t)i * 16u);
    async_copy_b128(l2 + (uint32_t)i * 16u, g2, (uint32_t)i * 16u);
  }
  for (int i = tid; i < (NW * C) / 4; i += BLOCK) {   // 880 x b128
    async_copy_b128(lw + (uint32_t)i * 16u, gw, (uint32_t)i * 16u);
  }

  // ---- Overlapped with the DMA: pack COO metadata into LDS, zero s_out ----
  // m1,m2,mo in [0,25): 5 bits; l in [0,55): 6 bits.
  for (int k = tid; k < NNZ; k += BLOCK) {
    uint32_t pk = (uint32_t)m1[k] | ((uint32_t)m2[k] << 5) |
                  ((uint32_t)l_ind[k] << 10) | ((uint32_t)m_out[k] << 16);
    s_meta[k] = make_uint2(pk, __float_as_uint(cg[k]));
  }
  for (int i = tid; i < NM * C; i += BLOCK) s_out[i] = 0.0f;

  wait_asynccnt0();     // this wave's DMA data landed in LDS
  __syncthreads();      // everyone's data + meta visible

  // ---- Main loop: thread = (channel quad, contiguous k-chunk) -------------
  const int cq = tid & (QUADS - 1);
  const int g  = tid >> 4;            // wave-uniform per half-wave
  const int c0 = cq * 4;

  int k    = g * CHUNK;
  int kend = (k + CHUNK < NNZ) ? (k + CHUNK) : NNZ;

  float ax = 0.f, ay = 0.f, az = 0.f, aw = 0.f;
  int cur_mo = -1;

  for (; k < kend; ++k) {
    const uint2 mk = s_meta[k];                 // ds_load_b64, broadcast
    const uint32_t pk = mk.x;
    const float cgv = __uint_as_float(mk.y);
    const int i1 = (int)(pk & 31u);
    const int i2 = (int)((pk >> 5) & 31u);
    const int il = (int)((pk >> 10) & 63u);
    const int mo = (int)((pk >> 16) & 31u);

    const v4f a  = *(const v4f*)&s_x1[i1 * C + c0];   // ds_load_b128
    const v4f bb = *(const v4f*)&s_x2[i2 * C + c0];
    const v4f ww = *(const v4f*)&s_w [il * C + c0];

    if (mo != cur_mo) {                 // M_out sorted -> rare, uniform/halfwave
      if (cur_mo >= 0) {
        float* d = &s_out[cur_mo * C + c0];
        atomicAdd(d + 0, ax); atomicAdd(d + 1, ay);
        atomicAdd(d + 2, az); atomicAdd(d + 3, aw);
      }
      ax = ay = az = aw = 0.f;
      cur_mo = mo;
    }
    ax = fmaf(a.x * bb.x * ww.x, cgv, ax);
    ay = fmaf(a.y * bb.y * ww.y, cgv, ay);
    az = fmaf(a.z * bb.z * ww.z, cgv, az);
    aw = fmaf(a.w * bb.w * ww.w, cgv, aw);
  }
  if (cur_mo >= 0) {
    float* d = &s_out[cur_mo * C + c0];
    atomicAdd(d + 0, ax); atomicAdd(d + 1, ay);
    atomicAdd(d + 2, az); atomicAdd(d + 3, aw);
  }
  __syncthreads();

  // ---- Write result (write-once -> non-temporal b128 stores) --------------
  float* gout = out + (size_t)b * (NM * C);
  for (int i = tid; i < (NM * C) / 4; i += BLOCK) {
    const v4f v = *(const v4f*)&s_out[i * 4];
    __builtin_nontemporal_store(v, (v4f*)&gout[i * 4]);
  }
}

extern "C" void kernel_launch(void* const* d_in, const int* in_sizes, int n_in,
                              void* d_out, int out_size, void* d_ws, size_t ws_size,
                              hipStream_t stream) {
  (void)n_in; (void)d_ws; (void)ws_size; (void)out_size;
  const float* x1 = (const float*)d_in[0];
  const float* x2 = (const float*)d_in[1];
  const float* w  = (const float*)d_in[2];
  const float* cg = (const float*)d_in[3];
  const int* mo   = (const int*)d_in[4];
  const int* m1   = (const int*)d_in[5];
  const int* m2   = (const int*)d_in[6];
  const int* li   = (const int*)d_in[7];
  float* out      = (float*)d_out;

  const int B = in_sizes[0] / (cfg::NM * cfg::C);   // 4096
  wtp_kernel<<<dim3((unsigned)B), dim3(cfg::BLOCK), 0, stream>>>(
      x1, x2, w, cg, mo, m1, m2, li, out);
}